// Gemma3Attention_44727789421383
// MI455X (gfx1250) — compile-verified
//
#include <hip/hip_runtime.h>
#include <math.h>

#define HIDDEN 3840
#define NUM_HEADS 16
#define NUM_KV_HEADS 8
#define HEAD_DIM 256
#define WINDOW 1024
#define BATCH 32
#define QDIM (NUM_HEADS * HEAD_DIM)      /* 4096 */
#define KVDIM (NUM_KV_HEADS * HEAD_DIM)  /* 2048 */
#define ATTN_SCALE 0.0625f               /* 256^-0.5 */
#define EPS 1e-6f
#define LOG_ROPE_BASE 9.210340371976184f /* ln(10000) */
#define NCHUNK 8
#define CHUNK (WINDOW / NCHUNK)          /* 128 */
#define MASKED_SCORE -1e30f

typedef float v2f __attribute__((ext_vector_type(2)));
typedef float v4f __attribute__((ext_vector_type(4)));
typedef float v8f __attribute__((ext_vector_type(8)));

// ---------------------------------------------------------------------------
// C[32 x N] = A[32 x K] @ B[K x N], row-major f32.
// One wave per 16-column tile; two 16x16 f32 accumulators cover all 32 rows,
// so each weight element of B is streamed from HBM exactly once.
// V_WMMA_F32_16X16X4_F32: exact f32 GEMM on the matrix pipe.
// ---------------------------------------------------------------------------
__global__ __launch_bounds__(32)
void gemm32_f32_wmma(const float* __restrict__ A, const float* __restrict__ B,
                     float* __restrict__ C, int K, int N) {
  const int lane = threadIdx.x & 31;
  const int half = lane >> 4;   // 0: lanes 0-15 (K=0,1), 1: lanes 16-31 (K=2,3)
  const int lid  = lane & 15;
  const int n0   = blockIdx.x * 16;

  v8f acc0 = {};  // rows 0..15
  v8f acc1 = {};  // rows 16..31

  const float* arow0 = A + (size_t)lid * K;
  const float* arow1 = A + (size_t)(16 + lid) * K;

  for (int k0 = 0; k0 < K; k0 += 4) {
    const int ka = k0 + 2 * half;
    // Prefetch the B stream ahead (global_prefetch_b8; speculative, OOB-safe).
    __builtin_prefetch(&B[(size_t)(ka + 32) * N + n0 + lid], 0, 1);
    v2f a0, a1, b;
    a0.x = arow0[ka];     a0.y = arow0[ka + 1];
    a1.x = arow1[ka];     a1.y = arow1[ka + 1];
    b.x  = B[(size_t)ka * N + n0 + lid];
    b.y  = B[(size_t)(ka + 1) * N + n0 + lid];
    acc0 = __builtin_amdgcn_wmma_f32_16x16x4_f32(false, a0, false, b,
                                                 (short)0, acc0, false, false);
    acc1 = __builtin_amdgcn_wmma_f32_16x16x4_f32(false, a1, false, b,
                                                 (short)0, acc1, false, false);
  }

#pragma unroll
  for (int r = 0; r < 8; ++r) {
    C[(size_t)(r + 8 * half) * N + n0 + lid]      = acc0[r];
    C[(size_t)(16 + r + 8 * half) * N + n0 + lid] = acc1[r];
  }
}

// ---------------------------------------------------------------------------
// Per-head RMS-norm + RoPE, in place: blocks [0,512) = q heads (q_ws),
// [512,768) = k heads (k_ws). 256 threads = one per head dim.
// ---------------------------------------------------------------------------
__global__ __launch_bounds__(256)
void norm_rope(float* __restrict__ q_ws, float* __restrict__ k_ws,
               const float* __restrict__ qw, const float* __restrict__ kw,
               const int* __restrict__ pos_p) {
  __shared__ float red[256];
  __shared__ float vec[HEAD_DIM];
  const int blk = blockIdx.x;
  const int t = threadIdx.x;
  const float pos = (float)(*pos_p);

  float* vptr;
  const float* w;
  if (blk < 512) {
    const int b = blk >> 4, h = blk & 15;
    vptr = q_ws + (size_t)b * QDIM + h * HEAD_DIM;
    w = qw;
  } else {
    const int i = blk - 512, b = i >> 3, h = i & 7;
    vptr = k_ws + (size_t)b * KVDIM + h * HEAD_DIM;
    w = kw;
  }

  const float x = vptr[t];
  red[t] = x * x;
  __syncthreads();
#pragma unroll
  for (int s = 128; s > 0; s >>= 1) {
    if (t < s) red[t] += red[t + s];
    __syncthreads();
  }
  const float rms = rsqrtf(red[0] * (1.0f / HEAD_DIM) + EPS);
  const float n = x * rms * (1.0f + w[t]);
  vec[t] = n;
  __syncthreads();

  const int j = (t < 128) ? t : (t - 128);
  const float theta = pos * __expf(-((float)j * (1.0f / 128.0f)) * LOG_ROPE_BASE);
  const float c = __cosf(theta), s = __sinf(theta);
  vptr[t] = (t < 128) ? (n * c - vec[t + 128] * s)
                      : (n * c + vec[t - 128] * s);
}

// ---------------------------------------------------------------------------
// Fused cache-copy + split-window attention (phase 1).
// Block = (b, kvh, chunk of 128 slots), 256 threads (8 waves).
// Each K/V row is read from HBM exactly ONCE: it is written to the output
// cache (nontemporal, single-use) and consumed for attention in the same
// pass. The write_idx slot sources the freshly roped k / new v instead.
// Produces per-chunk partial outputs + (max, sum) for flash-style merge.
// ---------------------------------------------------------------------------
__global__ __launch_bounds__(256)
void cache_attn_phase1(const float* __restrict__ kc_in, const float* __restrict__ vc_in,
                       const float* __restrict__ knew, const float* __restrict__ vnew,
                       const float* __restrict__ q_ws,
                       float* __restrict__ kc_out, float* __restrict__ vc_out,
                       float* __restrict__ pacc, float* __restrict__ pml,
                       const int* __restrict__ ci_p) {
  __shared__ float sc0[CHUNK];
  __shared__ float sc1[CHUNK];
  __shared__ float red[256];

  const int blk = blockIdx.x;
  const int chunk = blk & (NCHUNK - 1);
  const int kvh = (blk >> 3) & 7;
  const int b = blk >> 6;
  const int t = threadIdx.x, wave = t >> 5, lane = t & 31;
  const int s0 = chunk * CHUNK;
  const int ci = *ci_p;
  const int widx = ci % WINDOW;
  const int valid = min(ci + 1, WINDOW);

  const int h0 = kvh * 2, h1 = h0 + 1;
  const float* q0p = q_ws + (size_t)b * QDIM + h0 * HEAD_DIM + lane * 8;
  const float* q1p = q_ws + (size_t)b * QDIM + h1 * HEAD_DIM + lane * 8;
  const v4f qa0 = *(const v4f*)(q0p);
  const v4f qa1 = *(const v4f*)(q0p + 4);
  const v4f qb0 = *(const v4f*)(q1p);
  const v4f qb1 = *(const v4f*)(q1p + 4);

  const size_t head_base = ((size_t)(b * NUM_KV_HEADS + kvh)) * WINDOW * HEAD_DIM;
  const float* knew_row = knew + (size_t)b * KVDIM + kvh * HEAD_DIM;
  const float* vnew_row = vnew + (size_t)b * KVDIM + kvh * HEAD_DIM;

  // --- K pass: copy-out + scores, one wave per 16 slots ---
  for (int i = 0; i < CHUNK / 8; ++i) {
    const int s = s0 + wave * (CHUNK / 8) + i;
    const float* row = (s == widx) ? knew_row : (kc_in + head_base + (size_t)s * HEAD_DIM);
    const v4f k0 = __builtin_nontemporal_load((const v4f*)(row + lane * 8));
    const v4f k1 = __builtin_nontemporal_load((const v4f*)(row + lane * 8 + 4));
    float* orow = kc_out + head_base + (size_t)s * HEAD_DIM + lane * 8;
    __builtin_nontemporal_store(k0, (v4f*)(orow));
    __builtin_nontemporal_store(k1, (v4f*)(orow + 4));
    float d0 = qa0.x * k0.x + qa0.y * k0.y + qa0.z * k0.z + qa0.w * k0.w +
               qa1.x * k1.x + qa1.y * k1.y + qa1.z * k1.z + qa1.w * k1.w;
    float d1 = qb0.x * k0.x + qb0.y * k0.y + qb0.z * k0.z + qb0.w * k0.w +
               qb1.x * k1.x + qb1.y * k1.y + qb1.z * k1.z + qb1.w * k1.w;
#pragma unroll
    for (int off = 16; off > 0; off >>= 1) {
      d0 += __shfl_xor(d0, off, 32);
      d1 += __shfl_xor(d1, off, 32);
    }
    if (lane == 0) {
      const bool ok = (s < valid);
      sc0[s - s0] = ok ? d0 * ATTN_SCALE : MASKED_SCORE;
      sc1[s - s0] = ok ? d1 * ATTN_SCALE : MASKED_SCORE;
    }
  }
  __syncthreads();

  // --- chunk-local max (threads 0-127: head0, 128-255: head1) ---
  red[t] = (t < CHUNK) ? sc0[t] : sc1[t - CHUNK];
  __syncthreads();
#pragma unroll
  for (int s = 64; s > 0; s >>= 1) {
    if ((t & 127) < s) red[t] = fmaxf(red[t], red[t + s]);
    __syncthreads();
  }
  const float m0 = red[0], m1 = red[128];
  __syncthreads();

  // --- exp + chunk-local sum ---
  float e;
  if (t < CHUNK) { e = __expf(sc0[t] - m0); sc0[t] = e; }
  else           { e = __expf(sc1[t - CHUNK] - m1); sc1[t - CHUNK] = e; }
  red[t] = e;
  __syncthreads();
#pragma unroll
  for (int s = 64; s > 0; s >>= 1) {
    if ((t & 127) < s) red[t] += red[t + s];
    __syncthreads();
  }
  const float l0 = red[0], l1 = red[128];

  // --- V pass: copy-out + partial weighted accumulation (coalesced) ---
  float acc0 = 0.f, acc1 = 0.f;
  for (int i = 0; i < CHUNK; ++i) {
    const int s = s0 + i;
    const float* row = (s == widx) ? vnew_row : (vc_in + head_base + (size_t)s * HEAD_DIM);
    const float v = __builtin_nontemporal_load(row + t);
    __builtin_nontemporal_store(v, vc_out + head_base + (size_t)s * HEAD_DIM + t);
    acc0 += sc0[i] * v;
    acc1 += sc1[i] * v;
  }

  // --- store partials ---
  const size_t p0 = ((size_t)(b * NUM_HEADS + h0) * NCHUNK + chunk) * HEAD_DIM;
  const size_t p1 = ((size_t)(b * NUM_HEADS + h1) * NCHUNK + chunk) * HEAD_DIM;
  pacc[p0 + t] = acc0;
  pacc[p1 + t] = acc1;
  if (t == 0) {
    const size_t q0i = ((size_t)(b * NUM_HEADS + h0) * NCHUNK + chunk) * 2;
    const size_t q1i = ((size_t)(b * NUM_HEADS + h1) * NCHUNK + chunk) * 2;
    pml[q0i] = m0; pml[q0i + 1] = l0;
    pml[q1i] = m1; pml[q1i + 1] = l1;
  }
}

// ---------------------------------------------------------------------------
// Phase 2: merge the NCHUNK partials per (b,h) with max-rescaling.
// ---------------------------------------------------------------------------
__global__ __launch_bounds__(256)
void attn_combine(const float* __restrict__ pacc, const float* __restrict__ pml,
                  float* __restrict__ attn_out) {
  const int bh = blockIdx.x;  // b*16 + h
  const int t = threadIdx.x;

  float m[NCHUNK], l[NCHUNK];
  float M = MASKED_SCORE;
#pragma unroll
  for (int c = 0; c < NCHUNK; ++c) {
    m[c] = pml[((size_t)bh * NCHUNK + c) * 2];
    l[c] = pml[((size_t)bh * NCHUNK + c) * 2 + 1];
    M = fmaxf(M, m[c]);
  }
  float denom = 0.f, out = 0.f;
#pragma unroll
  for (int c = 0; c < NCHUNK; ++c) {
    const float r = __expf(m[c] - M);
    denom += r * l[c];
    out += r * pacc[((size_t)bh * NCHUNK + c) * HEAD_DIM + t];
  }
  attn_out[(size_t)bh * HEAD_DIM + t] = out / denom;
}

// ---------------------------------------------------------------------------
extern "C" void kernel_launch(void* const* d_in, const int* in_sizes, int n_in,
                              void* d_out, int out_size, void* d_ws, size_t ws_size,
                              hipStream_t stream) {
  (void)in_sizes; (void)n_in; (void)out_size; (void)ws_size;
  const float* x    = (const float*)d_in[0];
  const float* Wq   = (const float*)d_in[1];
  const float* Wk   = (const float*)d_in[2];
  const float* Wv   = (const float*)d_in[3];
  const float* Wo   = (const float*)d_in[4];
  const float* qnw  = (const float*)d_in[5];
  const float* knw  = (const float*)d_in[6];
  const float* kc_i = (const float*)d_in[7];
  const float* vc_i = (const float*)d_in[8];
  const int*   posp = (const int*)d_in[9];
  const int*   cip  = (const int*)d_in[10];

  const size_t cache_elems = (size_t)BATCH * NUM_KV_HEADS * WINDOW * HEAD_DIM;
  float* y_out  = (float*)d_out;                        // 32*3840
  float* kc_out = y_out + (size_t)BATCH * HIDDEN;       // updated k cache
  float* vc_out = kc_out + cache_elems;                 // updated v cache

  float* q_ws  = (float*)d_ws;                          // 32*4096
  float* k_ws  = q_ws + (size_t)BATCH * QDIM;           // 32*2048
  float* v_ws  = k_ws + (size_t)BATCH * KVDIM;          // 32*2048
  float* a_ws  = v_ws + (size_t)BATCH * KVDIM;          // 32*4096
  float* pacc  = a_ws + (size_t)BATCH * QDIM;           // 512*8*256
  float* pml   = pacc + (size_t)BATCH * NUM_HEADS * NCHUNK * HEAD_DIM;  // 512*8*2

  // QKV projections (f32 WMMA GEMMs, weights streamed once).
  gemm32_f32_wmma<<<QDIM / 16, 32, 0, stream>>>(x, Wq, q_ws, HIDDEN, QDIM);
  gemm32_f32_wmma<<<KVDIM / 16, 32, 0, stream>>>(x, Wk, k_ws, HIDDEN, KVDIM);
  gemm32_f32_wmma<<<KVDIM / 16, 32, 0, stream>>>(x, Wv, v_ws, HIDDEN, KVDIM);

  // RMS-norm + RoPE for q and k (in place).
  norm_rope<<<768, 256, 0, stream>>>(q_ws, k_ws, qnw, knw, posp);

  // Fused cache-copy + split-window attention: caches read once, written once.
  cache_attn_phase1<<<BATCH * NUM_KV_HEADS * NCHUNK, 256, 0, stream>>>(
      kc_i, vc_i, k_ws, v_ws, q_ws, kc_out, vc_out, pacc, pml, cip);

  // Merge chunk partials.
  attn_combine<<<BATCH * NUM_HEADS, 256, 0, stream>>>(pacc, pml, a_ws);

  // Output projection.
  gemm32_f32_wmma<<<HIDDEN / 16, 32, 0, stream>>>(a_ws, Wo, y_out, QDIM, HIDDEN);
}